// LSDANLayer_23210003268193
// MI455X (gfx1250) — compile-verified
//
#include <hip/hip_runtime.h>
#include <hip/hip_bf16.h>
#include <math.h>

#define NNODES  100000
#define NEDGES  1600000
#define KHOPS   3
#define DIM     128
#define NTILES  (NNODES / 16)   // 6250, exact

typedef __attribute__((ext_vector_type(2))) float v2f;
typedef __attribute__((ext_vector_type(8))) float v8f;

// One wave computes a 16x128 f32 tile: D = A(16x128) * B(128x128 cols [0..127]).
// Uses V_WMMA_F32_16X16X4_F32, K stepped by 4.
// A frag (16x4 f32): lanes 0-15 hold M=lane, K={k0,k0+1}; lanes 16-31 hold K={k0+2,k0+3}.
// B frag (4x16 f32): lanes 0-15 hold N=lane, K rows {k0,k0+1}; lanes 16-31 K rows {k0+2,k0+3}.
__device__ __forceinline__ void tile_gemm16x128(const float* __restrict__ arow,
                                                const float* __restrict__ B,
                                                int half, int m, v8f acc[8]) {
    for (int k0 = 0; k0 < DIM; k0 += 4) {
        const int ka = k0 + 2 * half;
        v2f a = { arow[ka], arow[ka + 1] };
#pragma unroll
        for (int t = 0; t < 8; ++t) {
            const int col = t * 16 + m;
            v2f b = { B[ka * DIM + col], B[ka * DIM + DIM + col] };
            acc[t] = __builtin_amdgcn_wmma_f32_16x16x4_f32(
                false, a, false, b, (short)0, acc[t], false, false);
        }
    }
}

// ---- Kernel 1: h = x @ W_short, plus fused a_i = h@w[:D], a_j = h@w[D:] ----
__global__ void __launch_bounds__(256)
k_gemm_short(const float* __restrict__ x, const float* __restrict__ W,
             const float* __restrict__ attw,
             float* __restrict__ h, float* __restrict__ a_i, float* __restrict__ a_j) {
    const int wave = (blockIdx.x * blockDim.x + threadIdx.x) >> 5;
    const int lane = threadIdx.x & 31;
    if (wave >= NTILES) return;                   // whole-wave uniform: EXEC all-1 for WMMA
    const int row0 = wave * 16;
    const int half = lane >> 4;
    const int m    = lane & 15;

    v8f acc[8];
    const v8f zero = {0.f,0.f,0.f,0.f,0.f,0.f,0.f,0.f};
#pragma unroll
    for (int t = 0; t < 8; ++t) acc[t] = zero;

    tile_gemm16x128(x + (size_t)(row0 + m) * DIM, W, half, m, acc);

    // column weights for the two attention halves (per-lane column = t*16+m)
    float w1[8], w2[8];
#pragma unroll
    for (int t = 0; t < 8; ++t) { w1[t] = attw[t * 16 + m]; w2[t] = attw[DIM + t * 16 + m]; }

    // C/D layout: acc[t][r] = D[M = r + 8*half, N = t*16 + m]
#pragma unroll
    for (int r = 0; r < 8; ++r) {
        const int row = row0 + r + 8 * half;
        float si = 0.f, sj = 0.f;
#pragma unroll
        for (int t = 0; t < 8; ++t) {
            const float v = acc[t][r];
            h[(size_t)row * DIM + t * 16 + m] = v;
            si += v * w1[t];
            sj += v * w2[t];
        }
        // reduce across the 16 lanes of this half-wave
#pragma unroll
        for (int off = 8; off >= 1; off >>= 1) {
            si += __shfl_xor(si, off, 32);
            sj += __shfl_xor(sj, off, 32);
        }
        if (m == 0) { a_i[row] = si; a_j[row] = sj; }
    }
}

// ---- Kernel 2: short-path edge scatter. One wave per edge, float4 per lane ----
__global__ void __launch_bounds__(256)
k_short_edges(const int* __restrict__ ei, const float* __restrict__ h,
              const float* __restrict__ a_i, const float* __restrict__ a_j,
              const float* __restrict__ attb, float* __restrict__ out) {
    const int wave = (blockIdx.x * blockDim.x + threadIdx.x) >> 5;
    const int lane = threadIdx.x & 31;
    if (wave >= NEDGES) return;
    const int src = ei[wave];
    const int dst = ei[NEDGES + wave];
    float s = a_i[dst] + a_j[src] + attb[0];
    s = (s >= 0.f) ? s : 0.2f * s;
    s = __expf(s);
    const float4 v = ((const float4*)(h + (size_t)src * DIM))[lane];
    float* o = out + (size_t)dst * DIM + lane * 4;
    unsafeAtomicAdd(o + 0, s * v.x);
    unsafeAtomicAdd(o + 1, s * v.y);
    unsafeAtomicAdd(o + 2, s * v.z);
    unsafeAtomicAdd(o + 3, s * v.w);
}

// ---- Kernel 3: per-hop SpMM scatter: agg[k][dst] += val * x[src] ----
__global__ void __launch_bounds__(256)
k_hop_edges(const int* __restrict__ hsrc, const int* __restrict__ hdst,
            const float* __restrict__ hval, const float* __restrict__ x,
            float* __restrict__ agg) {
    const int wave = (blockIdx.x * blockDim.x + threadIdx.x) >> 5;
    const int lane = threadIdx.x & 31;
    const int k    = blockIdx.y;
    if (wave >= NEDGES) return;
    const size_t off = (size_t)k * NEDGES + wave;
    const int src = hsrc[off];
    const int dst = hdst[off];
    const float v = hval[off];
    const float4 xv = ((const float4*)(x + (size_t)src * DIM))[lane];
    float* o = agg + ((size_t)k * NNODES + dst) * DIM + lane * 4;
    unsafeAtomicAdd(o + 0, v * xv.x);
    unsafeAtomicAdd(o + 1, v * xv.y);
    unsafeAtomicAdd(o + 2, v * xv.z);
    unsafeAtomicAdd(o + 3, v * xv.w);
}

// ---- Kernel 4: fused long path: hk_k = leaky(agg_k @ W_long), hop softmax,
//      weighted sum; adds onto d_out (which already holds the short result). ----
__global__ void __launch_bounds__(256)
k_long_fused(const float* __restrict__ agg, const float* __restrict__ Wl,
             const float* __restrict__ attw, const float* __restrict__ attb,
             float* __restrict__ out) {
    const int wave = (blockIdx.x * blockDim.x + threadIdx.x) >> 5;
    const int lane = threadIdx.x & 31;
    if (wave >= NTILES) return;
    const int row0 = wave * 16;
    const int half = lane >> 4;
    const int m    = lane & 15;
    const v8f zero = {0.f,0.f,0.f,0.f,0.f,0.f,0.f,0.f};

    float wcol[8];
#pragma unroll
    for (int t = 0; t < 8; ++t) wcol[t] = attw[t * 16 + m];
    const float b0 = attb[0];

    // Pass 1: per-hop logits (GEMM + leaky + row-dot + 16-lane reduce)
    float lg[KHOPS][8];
#pragma unroll
    for (int k = 0; k < KHOPS; ++k) {
        const float* arow = agg + ((size_t)k * NNODES + row0 + m) * DIM;
        v8f acc[8];
#pragma unroll
        for (int t = 0; t < 8; ++t) acc[t] = zero;
        tile_gemm16x128(arow, Wl, half, m, acc);
#pragma unroll
        for (int r = 0; r < 8; ++r) {
            float s = 0.f;
#pragma unroll
            for (int t = 0; t < 8; ++t) {
                float v = acc[t][r];
                v = (v >= 0.f) ? v : 0.01f * v;
                s += v * wcol[t];
            }
#pragma unroll
            for (int off = 8; off >= 1; off >>= 1) s += __shfl_xor(s, off, 32);
            lg[k][r] = s + b0;
        }
    }

    // Hop softmax per row
    float at[KHOPS][8];
#pragma unroll
    for (int r = 0; r < 8; ++r) {
        const float mx = fmaxf(lg[0][r], fmaxf(lg[1][r], lg[2][r]));
        const float e0 = __expf(lg[0][r] - mx);
        const float e1 = __expf(lg[1][r] - mx);
        const float e2 = __expf(lg[2][r] - mx);
        const float inv = 1.f / (e0 + e1 + e2);
        at[0][r] = e0 * inv; at[1][r] = e1 * inv; at[2][r] = e2 * inv;
    }

    // Pass 2: recompute hk_k, accumulate attn-weighted sum (cheaper than holding
    // 3 hops × 64 accumulator VGPRs live simultaneously)
    v8f oacc[8];
#pragma unroll
    for (int t = 0; t < 8; ++t) oacc[t] = zero;
#pragma unroll
    for (int k = 0; k < KHOPS; ++k) {
        const float* arow = agg + ((size_t)k * NNODES + row0 + m) * DIM;
        v8f acc[8];
#pragma unroll
        for (int t = 0; t < 8; ++t) acc[t] = zero;
        tile_gemm16x128(arow, Wl, half, m, acc);
#pragma unroll
        for (int t = 0; t < 8; ++t)
#pragma unroll
            for (int r = 0; r < 8; ++r) {
                float v = acc[t][r];
                v = (v >= 0.f) ? v : 0.01f * v;
                oacc[t][r] += at[k][r] * v;
            }
    }

    // out += long (short already accumulated; stream-ordered, no races per tile)
#pragma unroll
    for (int r = 0; r < 8; ++r) {
        const size_t row = row0 + r + 8 * half;
#pragma unroll
        for (int t = 0; t < 8; ++t) {
            const size_t idx = row * DIM + t * 16 + m;
            out[idx] = out[idx] + oacc[t][r];
        }
    }
}

extern "C" void kernel_launch(void* const* d_in, const int* in_sizes, int n_in,
                              void* d_out, int out_size, void* d_ws, size_t ws_size,
                              hipStream_t stream) {
    const float* x           = (const float*)d_in[0];
    const int*   edge_index  = (const int*)d_in[1];
    const int*   hop_src     = (const int*)d_in[2];
    const int*   hop_dst     = (const int*)d_in[3];
    const float* hop_vals    = (const float*)d_in[4];
    const float* W_short     = (const float*)d_in[5];
    const float* att_short_w = (const float*)d_in[6];
    const float* att_short_b = (const float*)d_in[7];
    const float* W_long      = (const float*)d_in[8];
    const float* att_long_w  = (const float*)d_in[9];
    const float* att_long_b  = (const float*)d_in[10];
    float* out = (float*)d_out;

    // workspace layout (floats): h[N*128] | a_i[N] | a_j[N] | agg[3*N*128]
    float* ws  = (float*)d_ws;
    float* h   = ws;
    float* a_i = h + (size_t)NNODES * DIM;
    float* a_j = a_i + NNODES;
    float* agg = a_j + NNODES;

    hipMemsetAsync(out, 0, (size_t)NNODES * DIM * sizeof(float), stream);
    hipMemsetAsync(agg, 0, (size_t)KHOPS * NNODES * DIM * sizeof(float), stream);

    const int gemm_blocks = (NTILES * 32 + 255) / 256;   // one wave32 per 16-row tile
    const int edge_blocks = NEDGES / 8;                  // one wave32 per edge, 8 waves/block

    k_gemm_short<<<dim3(gemm_blocks), dim3(256), 0, stream>>>(
        x, W_short, att_short_w, h, a_i, a_j);
    k_short_edges<<<dim3(edge_blocks), dim3(256), 0, stream>>>(
        edge_index, h, a_i, a_j, att_short_b, out);
    k_hop_edges<<<dim3(edge_blocks, KHOPS), dim3(256), 0, stream>>>(
        hop_src, hop_dst, hop_vals, x, agg);
    k_long_fused<<<dim3(gemm_blocks), dim3(256), 0, stream>>>(
        agg, W_long, att_long_w, att_long_b, out);
}